// NodeAggregator_5823975653925
// MI455X (gfx1250) — compile-verified
//
#include <hip/hip_runtime.h>
#include <stdint.h>

#define BATCH   4
#define N_NODES 64
#define TIME_   32
#define CH      128
#define EDGES   (N_NODES * (N_NODES - 1))   // 4032 edges

// ---------------------------------------------------------------------------
// NodeAggregator: out[b,e,t,0:128]   = node_msg[b, src[e], t, :]
//                 out[b,e,t,128:256] = node_msg[b, tgt[e], t, :]
// Pure data movement: 528 MB stores, 4 MB hot input (lives in 192 MB L2).
// Store-bandwidth bound (~23 us floor at 23.3 TB/s). CDNA5 async global<->LDS
// DMA path (ASYNCcnt): bytes flow HBM->LDS->HBM with no VGPR round trip and
// 8 outstanding B128 ops per wave. Stores carry TH=NT (cpol=1) so the 528 MB
// write stream does not evict the hot 4 MB input from L2; loads stay RT.
// ---------------------------------------------------------------------------

#if defined(__HIP_DEVICE_COMPILE__)

// Builtin signature (from hipcc diagnostic): pointee type is an int4 vector;
// global arg is in the device (AS1) address space, LDS side in AS3.
typedef int v4i_t __attribute__((ext_vector_type(4)));
typedef __attribute__((address_space(1))) v4i_t gv4i;
typedef __attribute__((address_space(3))) v4i_t lv4i;

__device__ __forceinline__ gv4i* as_global_v4(const void* p) {
  return (gv4i*)(uintptr_t)p;              // generic global addr == AS1 addr
}
__device__ __forceinline__ lv4i* as_lds_v4(const void* p) {
  // Generic LDS pointer: addr[31:0] is the LDS byte offset (aperture mapping).
  return (lv4i*)(uint32_t)(uintptr_t)p;
}

__device__ __forceinline__ void async_load_b128(const void* gsrc, void* ldst) {
#if __has_builtin(__builtin_amdgcn_global_load_async_to_lds_b128)
  __builtin_amdgcn_global_load_async_to_lds_b128(as_global_v4(gsrc), as_lds_v4(ldst),
                                                 /*offset=*/0, /*cpol=RT*/0);
#else
  uint32_t lofs  = (uint32_t)(uintptr_t)ldst;
  uint64_t gaddr = (uint64_t)(uintptr_t)gsrc;
  asm volatile("global_load_async_to_lds_b128 %0, %1, off"
               :: "v"(lofs), "v"(gaddr) : "memory");
#endif
}

__device__ __forceinline__ void async_store_b128(void* gdst, const void* lsrc) {
#if __has_builtin(__builtin_amdgcn_global_store_async_from_lds_b128)
  // cpol = 1 -> TH_NT: non-temporal streaming store (write-once output).
  __builtin_amdgcn_global_store_async_from_lds_b128(as_global_v4(gdst), as_lds_v4(lsrc),
                                                    /*offset=*/0, /*cpol=NT*/1);
#else
  uint32_t lofs  = (uint32_t)(uintptr_t)lsrc;
  uint64_t gaddr = (uint64_t)(uintptr_t)gdst;
  asm volatile("global_store_async_from_lds_b128 %0, %1, off th:TH_STORE_NT"
               :: "v"(gaddr), "v"(lofs) : "memory");
#endif
}

__device__ __forceinline__ void wait_async0() {
#if __has_builtin(__builtin_amdgcn_s_wait_asynccnt)
  __builtin_amdgcn_s_wait_asynccnt(0);
#else
  asm volatile("s_wait_asynccnt 0x0" ::: "memory");
#endif
}

#endif // __HIP_DEVICE_COMPILE__

__global__ __launch_bounds__(256) void NodeAggregator_gather_concat(
    const float* __restrict__ node_msg,
    const int*   __restrict__ esrc,
    const int*   __restrict__ etgt,
    float*       __restrict__ out)
{
#if defined(__HIP_DEVICE_COMPILE__)
  // 32 KB bounce buffer: [0,16KB) = source-node row, [16KB,32KB) = target row.
  __shared__ float lds[TIME_ * 2 * CH];

  const int e   = blockIdx.x;     // edge   (0..4031)
  const int b   = blockIdx.y;     // batch  (0..3)
  const int tid = threadIdx.x;    // 0..255  (8 wave32)

  const int ns = esrc[e];         // uniform per block -> scalar loads
  const int nt = etgt[e];

  const float* srcRow = node_msg + (((size_t)b * N_NODES + ns) * TIME_) * CH; // 16KB contiguous
  const float* tgtRow = node_msg + (((size_t)b * N_NODES + nt) * TIME_) * CH; // 16KB contiguous
  float*       outB   = out + (((size_t)b * EDGES + e) * TIME_) * (2 * CH);

  // 2048 float4 (32 KB) per block; lane handles j = tid + 256k, k = 0..7.
  // k<4 -> source half (j in [0,1024)), k>=4 -> target half.
  // Each wave's store set reads exactly the LDS bytes that wave loaded, so a
  // per-wave s_wait_asynccnt(0) suffices -- no workgroup barrier.

  // Phase 1: HBM/L2 -> LDS (ASYNCcnt, fire-and-forget; input fully contiguous)
#pragma unroll
  for (int k = 0; k < 8; ++k) {
    const int j  = tid + k * 256;                 // float4 index in [0,2048)
    const int jj = j & 1023;                      // float4 index within half
    const float* gin = ((k < 4) ? srcRow : tgtRow) + (size_t)jj * 4;
    async_load_b128(gin, &lds[(size_t)j * 4]);
  }

  wait_async0();   // this wave's LDS contents are resident

  // Phase 2: LDS -> HBM. Output row-half = 512B contiguous, 1024B stride;
  // one wave instruction covers exactly one 512B chunk (perfectly coalesced).
#pragma unroll
  for (int k = 0; k < 8; ++k) {
    const int j  = tid + k * 256;
    const int h  = j >> 10;        // 0 = source half, 1 = target half
    const int jj = j & 1023;
    const int t  = jj >> 5;        // time  (32 float4 per 128-ch row)
    const int c4 = jj & 31;        // float4 within 128 channels
    float* gout = outB + (size_t)t * (2 * CH) + h * CH + c4 * 4;
    async_store_b128(gout, &lds[(size_t)j * 4]);
  }
  // S_ENDPGM performs an implicit wait-idle: async stores drain before retire.
#endif
}

extern "C" void kernel_launch(void* const* d_in, const int* in_sizes, int n_in,
                              void* d_out, int out_size, void* d_ws, size_t ws_size,
                              hipStream_t stream) {
  const float* node_msg = (const float*)d_in[0];
  const int*   esrc     = (const int*)d_in[1];
  const int*   etgt     = (const int*)d_in[2];
  float*       out      = (float*)d_out;

  dim3 grid(EDGES, BATCH);   // 4032 x 4 blocks, 256 threads (8 wave32) each
  NodeAggregator_gather_concat<<<grid, 256, 0, stream>>>(node_msg, esrc, etgt, out);
}